// GIN_39376260170206
// MI455X (gfx1250) — compile-verified
//
#include <hip/hip_runtime.h>
#include <math.h>

#define D 128

typedef __attribute__((ext_vector_type(2))) float v2f;
typedef __attribute__((ext_vector_type(4))) float v4f;
typedef __attribute__((ext_vector_type(8))) float v8f;

// ---------------------------------------------------------------- zero aggr
__global__ __launch_bounds__(256) void gin_zero(float* __restrict__ p, long n4) {
    long i = (long)blockIdx.x * blockDim.x + threadIdx.x;
    if (i < n4) ((v4f*)p)[i] = (v4f){0.f, 0.f, 0.f, 0.f};
}

// ------------------------------------------------------ edge gather+scatter
// One wave32 per edge; lane l moves floats [4l, 4l+4) of the 128-wide row.
// Gathers hit L2 (node array = 51MB < 192MB L2); scatter via f32 atomics.
__global__ __launch_bounds__(256) void gin_scatter(const float* __restrict__ h,
                                                   const int* __restrict__ src,
                                                   const int* __restrict__ dst,
                                                   float* __restrict__ aggr, int E) {
    int tid = blockIdx.x * 256 + threadIdx.x;
    int e = tid >> 5;
    if (e >= E) return;
    int c = (tid & 31) << 2;
    int s = src[e];
    int d = dst[e];
    v4f v = *(const v4f*)(h + (size_t)s * D + c);
    float* p = aggr + (size_t)d * D + c;
    atomicAdd(p + 0, v.x);
    atomicAdd(p + 1, v.y);
    atomicAdd(p + 2, v.z);
    atomicAdd(p + 3, v.w);
}

// ------------------------------------- out = ELU((h + aggr) @ W + b), WMMA
// 8 waves per block; wave w owns rows [16*(8*bid+w), +16). W staged in LDS.
// fp32 WMMA 16x16x4: K-loop of 32 steps, 8 column tiles accumulated in v8f.
// Safe in-place (out may alias h): each row block is read only by the wave
// that writes it, and all its reads precede its stores.
__global__ __launch_bounds__(256) void gin_gemm_elu(const float* h,
                                                    const float* __restrict__ aggr,
                                                    const float* __restrict__ W,
                                                    const float* __restrict__ b,
                                                    float* out, int n_rows) {
    __shared__ __align__(16) float ldsW[D * D];  // 64 KB of the 320 KB WGP LDS

    const int t = threadIdx.x;
    // Stage W (row-major KxN = 128x128) into LDS, 16 float4s per thread.
    #pragma unroll
    for (int i = t; i < D * D / 4; i += 256)
        ((v4f*)ldsW)[i] = ((const v4f*)W)[i];
    __syncthreads();

    const int wave = t >> 5;
    const int lane = t & 31;
    const int row0 = (blockIdx.x * 8 + wave) * 16;
    if (row0 >= n_rows) return;

    const int m    = lane & 15;           // row within tile (A) / col within tile (B)
    const int koff = (lane >> 4) << 1;    // lanes 0-15 -> K+{0,1}; lanes 16-31 -> K+{2,3}

    const float* arow = h    + (size_t)(row0 + m) * D;
    const float* grow = aggr + (size_t)(row0 + m) * D;

    v8f c[8] = {};  // 8 column tiles of 16x16 f32 accumulators

    for (int kt = 0; kt < 32; ++kt) {
        const int k0 = kt * 4 + koff;
        // A fragment: (h + aggr)[row0+m][k0, k0+1]  (fuses the GIN update)
        v2f ha = *(const v2f*)(arow + k0);
        v2f ga = *(const v2f*)(grow + k0);
        v2f a = ha + ga;
        #pragma unroll
        for (int nt = 0; nt < 8; ++nt) {
            const int n = nt * 16 + m;
            v2f bf;
            bf.x = ldsW[k0 * D + n];
            bf.y = ldsW[(k0 + 1) * D + n];
            // v_wmma_f32_16x16x4_f32: (neg_a, A, neg_b, B, c_mod, C, reuse_a, reuse_b)
            c[nt] = __builtin_amdgcn_wmma_f32_16x16x4_f32(
                false, a, false, bf, (short)0, c[nt], false, false);
        }
    }

    // Epilogue: bias + ELU(alpha=1), store.
    // C/D layout: VGPR v -> M = v + (lane<16 ? 0 : 8), N = lane%16.
    const int rbase = row0 + ((lane >> 4) << 3);
    #pragma unroll
    for (int nt = 0; nt < 8; ++nt) {
        const int col = nt * 16 + m;
        const float bias = b[col];
        #pragma unroll
        for (int v = 0; v < 8; ++v) {
            float x = c[nt][v] + bias;
            float y = x > 0.f ? x : expm1f(x);
            out[(size_t)(rbase + v) * D + col] = y;
        }
    }
}

// ---------------------------------------------------------------- launcher
extern "C" void kernel_launch(void* const* d_in, const int* in_sizes, int n_in,
                              void* d_out, int out_size, void* d_ws, size_t ws_size,
                              hipStream_t stream) {
    const float* x  = (const float*)d_in[0];
    const int*   ei = (const int*)d_in[1];
    const float* W0 = (const float*)d_in[2];
    const float* b0 = (const float*)d_in[3];
    const float* W1 = (const float*)d_in[4];
    const float* b1 = (const float*)d_in[5];
    const float* W2 = (const float*)d_in[6];
    const float* b2 = (const float*)d_in[7];

    const int N = in_sizes[0] / D;          // 100000
    const int E = in_sizes[1] / 2;          // 1600000
    const int* src = ei;                    // edge_index row 0
    const int* dst = ei + E;                // edge_index row 1

    float* out  = (float*)d_out;            // N*D fp32; doubles as h buffer
    float* aggr = (float*)d_ws;             // N*D fp32 scratch

    const long n4 = (long)N * D / 4;
    const int zb = (int)((n4 + 255) / 256);
    const int sb = (int)(((long)E * 32 + 255) / 256);
    const int gb = (N / 16 + 7) / 8;

    // Layer 0: reads x (inputs never mutated), writes out.
    gin_zero<<<zb, 256, 0, stream>>>(aggr, n4);
    gin_scatter<<<sb, 256, 0, stream>>>(x, src, dst, aggr, E);
    gin_gemm_elu<<<gb, 256, 0, stream>>>(x, aggr, W0, b0, out, N);

    // Layer 1: in-place on out.
    gin_zero<<<zb, 256, 0, stream>>>(aggr, n4);
    gin_scatter<<<sb, 256, 0, stream>>>(out, src, dst, aggr, E);
    gin_gemm_elu<<<gb, 256, 0, stream>>>(out, aggr, W1, b1, out, N);

    // Layer 2: in-place on out.
    gin_zero<<<zb, 256, 0, stream>>>(aggr, n4);
    gin_scatter<<<sb, 256, 0, stream>>>(out, src, dst, aggr, E);
    gin_gemm_elu<<<gb, 256, 0, stream>>>(out, aggr, W2, b2, out, N);
}